// GradTTS_69827578298731
// MI455X (gfx1250) — compile-verified
//
#include <hip/hip_runtime.h>
#include <cstdint>

#define B_  16
#define F_  80
#define TX  384
#define TY  1728
#define OUT 256
#define NEGV (-1e9f)
#define LPCONST (-73.5150826f)   // -0.5*log(2*pi)*80

typedef float v2f __attribute__((ext_vector_type(2)));
typedef float v8f __attribute__((ext_vector_type(8)));

// ---------------------------------------------------------------------------
// 1) Row/col squared-norm terms:  melsq[b,t] = -0.5*sum_f mels^2,
//                                 musq [b,x] = -0.5*sum_f mu^2
// ---------------------------------------------------------------------------
__global__ void sq_kernel(const float* __restrict__ mu_x,
                          const float* __restrict__ mels,
                          float* __restrict__ melsq,
                          float* __restrict__ musq) {
    int i = blockIdx.x * blockDim.x + threadIdx.x;
    if (i < B_ * TY) {
        int b = i / TY, t = i % TY;
        const float* p = mels + (size_t)b * F_ * TY + t;
        float s = 0.f;
        #pragma unroll 8
        for (int k = 0; k < F_; ++k) { float v = p[(size_t)k * TY]; s += v * v; }
        melsq[i] = -0.5f * s;
    } else if (i < B_ * (TY + TX)) {
        int j = i - B_ * TY;
        int b = j / TX, x = j % TX;
        const float* p = mu_x + (size_t)b * F_ * TX + x;
        float s = 0.f;
        #pragma unroll 8
        for (int k = 0; k < F_; ++k) { float v = p[(size_t)k * TX]; s += v * v; }
        musq[j] = -0.5f * s;
    }
}

// ---------------------------------------------------------------------------
// 2) WMMA GEMM: lpT[b,t,x] = sum_k mels[b,k,t]*mu_x[b,k,x] + const
//               + melsq_term[t] + musq_term[x], masked to NEG.
//    A = 16x4 tile of mels (rows = t), B = 4x16 tile of mu_x (cols = x),
//    D = 16x16 f32 accumulator.  One wave per 16x16 tile, 8 waves / block.
// ---------------------------------------------------------------------------
__global__ __launch_bounds__(256) void lp_gemm_kernel(
        const float* __restrict__ mu_x, const float* __restrict__ mels,
        const float* __restrict__ melsq, const float* __restrict__ musq,
        const int* __restrict__ text_lens, const int* __restrict__ mel_lens,
        float* __restrict__ lpT) {
    const int TILES_X = TX / 16;                 // 24
    const int TILES_T = TY / 16;                 // 108
    int wave = (int)((blockIdx.x * blockDim.x + threadIdx.x) >> 5);
    int lane = threadIdx.x & 31;
    int ntiles = B_ * TILES_T * TILES_X;
    if (wave >= ntiles) return;
    int xt = wave % TILES_X;
    int tt = (wave / TILES_X) % TILES_T;
    int b  = wave / (TILES_X * TILES_T);
    int x0 = xt * 16, t0 = tt * 16;
    int hh = lane >> 4;          // half-wave select (K pairs 0,1 vs 2,3)
    int mn = lane & 15;          // row (A) / col (B) index within tile
    int tm = t0 + mn;            // A-matrix row  (t coordinate)
    int xn = x0 + mn;            // B-matrix col  (x coordinate)
    const float* melB = mels + (size_t)b * F_ * TY;
    const float* muB  = mu_x + (size_t)b * F_ * TX;

    v8f acc = {};
    #pragma unroll 4
    for (int k0 = 0; k0 < F_; k0 += 4) {
        int ka = k0 + 2 * hh;
        v2f a, bm;
        a[0]  = melB[(size_t)ka * TY + tm];
        a[1]  = melB[(size_t)(ka + 1) * TY + tm];
        bm[0] = muB[(size_t)ka * TX + xn];
        bm[1] = muB[(size_t)(ka + 1) * TX + xn];
        acc = __builtin_amdgcn_wmma_f32_16x16x4_f32(
                false, a, false, bm, (short)0, acc, false, false);
    }

    int tlen = text_lens[b], mlen = mel_lens[b];
    float xterm = musq[b * TX + xn];
    bool  xvalid = (xn < tlen);
    float* outB = lpT + (size_t)b * TX * TY;
    #pragma unroll
    for (int i = 0; i < 8; ++i) {
        int trow = t0 + i + 8 * hh;                 // C/D: vgpr i -> row i + 8*half
        float val = acc[i] + LPCONST + xterm + melsq[b * TY + trow];
        bool valid = xvalid && (trow < mlen);
        outB[(size_t)trow * TX + xn] = valid ? val : NEGV;  // lanes -> consecutive x
    }
}

// ---------------------------------------------------------------------------
// 3) Forward DP: one wave32 per batch, 12 rows per lane (x = lane*12 + r).
//    Barrier-free: shift-by-one is register-local except one shfl_up per step.
//    Emits cmp bits:  bit(x,y) = value[x,y-1] < value[x-1,y-1].
//    Bit layout: word j (j=0..11) of column y, bit position l <-> x = l*12 + j.
// ---------------------------------------------------------------------------
__global__ __launch_bounds__(32) void dp_forward_kernel(
        const float* __restrict__ lpT, uint32_t* __restrict__ bits) {
    int b = blockIdx.x, lane = threadIdx.x;
    float v[12];
    #pragma unroll
    for (int r = 0; r < 12; ++r) v[r] = (lane == 0 && r == 0) ? 0.0f : NEGV;

    const float* base = lpT + (size_t)b * TX * TY + lane * 12;
    uint32_t* bb = bits + (size_t)b * TY * 12;

    for (int y = 0; y < TY; ++y) {
        const float4* cp = (const float4*)(base + (size_t)y * TX);
        float4 l0 = cp[0], l1 = cp[1], l2 = cp[2];
        float lp[12] = { l0.x, l0.y, l0.z, l0.w,
                         l1.x, l1.y, l1.z, l1.w,
                         l2.x, l2.y, l2.z, l2.w };

        float up = __shfl_up(v[11], 1, 32);
        if (lane == 0) up = NEGV;
        float sh[12];
        sh[0] = up;
        #pragma unroll
        for (int r = 1; r < 12; ++r) sh[r] = v[r - 1];

        uint32_t w[12];
        #pragma unroll
        for (int r = 0; r < 12; ++r) {
            bool bit = !(lane == 0 && r == 0) && (v[r] < sh[r]);
            w[r] = (uint32_t)__ballot(bit);       // uniform across the wave
        }
        uint32_t myw = w[0];
        #pragma unroll
        for (int r = 1; r < 12; ++r) myw = (lane == r) ? w[r] : myw;
        if (lane < 12) bb[(size_t)y * 12 + lane] = myw;

        #pragma unroll
        for (int r = 0; r < 12; ++r) v[r] = lp[r] + fmaxf(v[r], sh[r]);
    }
}

// ---------------------------------------------------------------------------
// 4) Backtrack: sequential walk per batch, cmp bits staged via LDS in
//    192-column chunks.  Writes compact path_idx[b,y] (-1 = inactive column).
// ---------------------------------------------------------------------------
#define CHUNK 192
__global__ __launch_bounds__(256) void backtrack_kernel(
        const uint32_t* __restrict__ bits,
        const int* __restrict__ text_lens, const int* __restrict__ mel_lens,
        int* __restrict__ pathIdx) {
    int b = blockIdx.x, tid = threadIdx.x;
    __shared__ uint32_t sb[CHUNK * 12];
    __shared__ int sidx[CHUNK];
    __shared__ int sCarry;
    int t_x = text_lens[b], t_y = mel_lens[b];
    if (tid == 0) sCarry = t_x - 1;
    const uint32_t* bbase = bits + (size_t)b * TY * 12;
    __syncthreads();
    for (int c = TY / CHUNK - 1; c >= 0; --c) {
        int y0 = c * CHUNK;
        for (int i = tid; i < CHUNK * 12; i += 256) sb[i] = bbase[(size_t)y0 * 12 + i];
        __syncthreads();
        if (tid == 0) {
            int idx = sCarry;
            for (int yy = CHUNK - 1; yy >= 0; --yy) {
                int y = y0 + yy;
                bool active = y < t_y;
                sidx[yy] = active ? idx : -1;
                bool move = (idx == y) ||
                            ((sb[yy * 12 + (idx % 12)] >> (idx / 12)) & 1u);
                if (active && idx > 0 && y > 0 && move) --idx;
            }
            sCarry = idx;
        }
        __syncthreads();
        if (tid < CHUNK) pathIdx[b * TY + y0 + tid] = sidx[tid];
        __syncthreads();
    }
}

// ---------------------------------------------------------------------------
// 5) Expand one-hot path into the attn output region (overwrites lpT scratch).
// ---------------------------------------------------------------------------
__global__ void write_attn_kernel(const int* __restrict__ pathIdx,
                                  float* __restrict__ attn) {
    size_t i = (size_t)blockIdx.x * blockDim.x + threadIdx.x;
    size_t total = (size_t)B_ * TX * TY / 4;
    if (i >= total) return;
    size_t e = i * 4;
    int y = (int)(e % TY);
    size_t bx = e / TY;
    int x = (int)(bx % TX);
    int b = (int)(bx / TX);
    const int* pp = pathIdx + b * TY + y;
    float4 o;
    o.x = (pp[0] == x) ? 1.0f : 0.0f;
    o.y = (pp[1] == x) ? 1.0f : 0.0f;
    o.z = (pp[2] == x) ? 1.0f : 0.0f;
    o.w = (pp[3] == x) ? 1.0f : 0.0f;
    ((float4*)attn)[i] = o;
}

// ---------------------------------------------------------------------------
// 6) mu_y (path is one-hot -> gather), mels_cut, y_cut_mask.
// ---------------------------------------------------------------------------
__global__ void cut_kernel(const float* __restrict__ mu_x,
                           const float* __restrict__ mels,
                           const int* __restrict__ mel_lens,
                           const int* __restrict__ pathIdx,
                           float* __restrict__ mu_y,
                           float* __restrict__ mels_cut,
                           float* __restrict__ mask) {
    int i = blockIdx.x * blockDim.x + threadIdx.x;
    if (i >= B_ * F_ * OUT) return;
    int yc = i % OUT;
    int bf = i / OUT;
    int f = bf % F_, b = bf / F_;
    int t_y = mel_lens[b];
    int off = (t_y > OUT) ? ((t_y - OUT) / 2) : 0;
    int cutlen = (t_y < OUT) ? t_y : OUT;
    bool m = yc < cutlen;
    int p = pathIdx[b * TY + off + yc];
    mu_y[i]    = (m && p >= 0) ? mu_x[((size_t)b * F_ + f) * TX + p] : 0.0f;
    mels_cut[i] = m ? mels[((size_t)b * F_ + f) * TY + off + yc] : 0.0f;
    if (f == 0) mask[b * OUT + yc] = m ? 1.0f : 0.0f;
}

// ---------------------------------------------------------------------------
extern "C" void kernel_launch(void* const* d_in, const int* in_sizes, int n_in,
                              void* d_out, int out_size, void* d_ws, size_t ws_size,
                              hipStream_t stream) {
    const float* mu_x      = (const float*)d_in[0];
    const float* mels      = (const float*)d_in[1];
    const int*   text_lens = (const int*)d_in[2];
    const int*   mel_lens  = (const int*)d_in[3];

    float* out      = (float*)d_out;
    float* mu_y     = out;                                  // [B,F,OUT]
    float* attn     = mu_y + (size_t)B_ * F_ * OUT;         // [B,Tx,Ty] (also lpT scratch)
    float* mels_cut = attn + (size_t)B_ * TX * TY;          // [B,F,OUT]
    float* mask     = mels_cut + (size_t)B_ * F_ * OUT;     // [B,OUT]

    float*    melsq   = (float*)d_ws;                       // B*Ty
    float*    musq    = melsq + (size_t)B_ * TY;            // B*Tx
    uint32_t* bits    = (uint32_t*)(musq + (size_t)B_ * TX);// B*Ty*12
    int*      pathIdx = (int*)(bits + (size_t)B_ * TY * 12);// B*Ty

    // 1) squared-norm terms
    int nsq = B_ * (TY + TX);
    sq_kernel<<<(nsq + 255) / 256, 256, 0, stream>>>(mu_x, mels, melsq, musq);

    // 2) WMMA log-prior GEMM -> lpT in attn region (transposed [b,t,x])
    int ntiles = B_ * (TY / 16) * (TX / 16);
    lp_gemm_kernel<<<(ntiles + 7) / 8, 256, 0, stream>>>(
        mu_x, mels, melsq, musq, text_lens, mel_lens, attn);

    // 3) forward DP (one wave per batch, barrier-free)
    dp_forward_kernel<<<B_, 32, 0, stream>>>(attn, bits);

    // 4) backtrack -> compact path indices
    backtrack_kernel<<<B_, 256, 0, stream>>>(bits, text_lens, mel_lens, pathIdx);

    // 5) expand one-hot path (overwrites lpT scratch)
    size_t q = (size_t)B_ * TX * TY / 4;
    write_attn_kernel<<<(unsigned)((q + 255) / 256), 256, 0, stream>>>(pathIdx, attn);

    // 6) cut outputs
    int nc = B_ * F_ * OUT;
    cut_kernel<<<(nc + 255) / 256, 256, 0, stream>>>(
        mu_x, mels, mel_lens, pathIdx, mu_y, mels_cut, mask);
}